// SolveNewsvendor_48318382080055
// MI455X (gfx1250) — compile-verified
//
#include <hip/hip_runtime.h>
#include <hip/hip_bf16.h>

#define KDIM 64
#define NITER 30

typedef float v2f __attribute__((ext_vector_type(2)));
typedef float v8f __attribute__((ext_vector_type(8)));

// Exact fp32 wave-wide sum of 64 values (2 per lane) through the matrix unit:
// D = ones(16x4) @ B(4x16)  =>  D[m,n] = colsum_n(B). Sum of ALL 256 D entries
// over the wave = 16 * total(B), independent of how operands map to lanes.
__device__ __forceinline__ float wave_sum64(float x0, float x1) {
    v2f a; a[0] = 1.0f; a[1] = 1.0f;       // A = ones(16x4), layout-invariant
    v2f b; b[0] = x0;   b[1] = x1;          // B carries the 64 summands
    v8f c = {0.f,0.f,0.f,0.f,0.f,0.f,0.f,0.f};
    v8f d = __builtin_amdgcn_wmma_f32_16x16x4_f32(
        false, a, false, b, (short)0, c, false, false);
    float t = ((d[0] + d[1]) + (d[2] + d[3])) + ((d[4] + d[5]) + (d[6] + d[7]));
    t += __shfl_xor(t, 16, 32);
    t += __shfl_xor(t,  8, 32);
    t += __shfl_xor(t,  4, 32);
    t += __shfl_xor(t,  2, 32);
    t += __shfl_xor(t,  1, 32);
    return t * 0.0625f;                     // /16
}

__device__ __forceinline__ float wave_min32(float t) {
    t = fminf(t, __shfl_xor(t, 16, 32));
    t = fminf(t, __shfl_xor(t,  8, 32));
    t = fminf(t, __shfl_xor(t,  4, 32));
    t = fminf(t, __shfl_xor(t,  2, 32));
    t = fminf(t, __shfl_xor(t,  1, 32));
    return t;
}

__device__ __forceinline__ float stepratio(float v, float dv, float big) {
    return (dv < 0.0f) ? (-v / dv) : big;
}

// One wave32 per batch sample. n = 1+2K = 129 variables, m = 1+4K = 257
// constraints. Lane owns indices i in {lane, lane+32} of each K=64 block;
// the lone block-3 row for z0 is replicated identically on all lanes.
__global__ __launch_bounds__(256) void newsvendor_ipm_kernel(
    const float* __restrict__ y, const float* __restrict__ q_base,
    const float* __restrict__ p_base, const float* __restrict__ h,
    float* __restrict__ out, int Bsz)
{
    const int lane   = threadIdx.x & 31;
    const int sample = blockIdx.x * (blockDim.x >> 5) + (threadIdx.x >> 5);
    if (sample >= Bsz) return;   // wave-uniform: EXEC stays all-ones for WMMA

    const float* yb = y + sample * KDIM;

    const float Qd0 = q_base[0];
    const float p0  = p_base[0];
    const float h3z = h[2 * KDIM];

    float qb[2], qh[2], pb[2], ph[2], h1[2], h2[2], h3b[2], h3h[2];
    #pragma unroll
    for (int u = 0; u < 2; ++u) {
        const int i = lane + 32 * u;
        const float yv = yb[i];                 // per-sample diag scale
        qb[u] = q_base[1 + i] * yv;             // Qd for "b" block
        qh[u] = q_base[1 + KDIM + i] * yv;      // Qd for "h" block
        pb[u] = p_base[1 + i] * yv;
        ph[u] = p_base[1 + KDIM + i] * yv;
        h1[u]  = h[i];                          // = -d_i
        h2[u]  = h[KDIM + i];                   // = +d_i
        h3b[u] = h[2 * KDIM + 1 + i];           // 0
        h3h[u] = h[2 * KDIM + 1 + KDIM + i];    // 0
    }

    // IPM state: z = 0, s = 1, lam = 1
    float z0 = 0.f;
    float zb[2]  = {0.f, 0.f}, zh[2]  = {0.f, 0.f};
    float s1[2]  = {1.f, 1.f}, s2[2]  = {1.f, 1.f};
    float s3b[2] = {1.f, 1.f}, s3h[2] = {1.f, 1.f};
    float l1[2]  = {1.f, 1.f}, l2[2]  = {1.f, 1.f};
    float l3b[2] = {1.f, 1.f}, l3h[2] = {1.f, 1.f};
    float s3z = 1.f, l3z = 1.f;                 // block-3 row for z0 (replicated)

    const float big   = 1e10f;
    const float sigma = 0.1f;
    const float inv_m = 1.0f / (1.0f + 4.0f * KDIM);   // 1/257

    #pragma unroll 1
    for (int it = 0; it < NITER; ++it) {
        // ---- primal residuals r_p = Gz + s - h, and mu partials ----
        float rp1[2], rp2[2], rp3b[2], rp3h[2], mupart[2];
        #pragma unroll
        for (int u = 0; u < 2; ++u) {
            rp1[u]  = -z0 - zb[u] + s1[u] - h1[u];
            rp2[u]  =  z0 - zh[u] + s2[u] - h2[u];
            rp3b[u] = -zb[u] + s3b[u] - h3b[u];
            rp3h[u] = -zh[u] + s3h[u] - h3h[u];
            mupart[u] = s1[u]*l1[u] + s2[u]*l2[u] + s3b[u]*l3b[u] + s3h[u]*l3h[u];
        }
        const float rp3z = -z0 + s3z - h3z;
        const float smu  = wave_sum64(mupart[0], mupart[1]) + s3z * l3z;
        const float sig_mu = sigma * smu * inv_m;

        // ---- centering r_c, w = (r_c - lam*r_p)/s, D = lam/s ----
        float rc1[2], rc2[2], rc3b[2], rc3h[2];
        float w1[2], w2[2], w3b[2], w3h[2];
        float D1[2], D2[2], D3b[2], D3h[2];
        #pragma unroll
        for (int u = 0; u < 2; ++u) {
            rc1[u]  = s1[u]*l1[u]   - sig_mu;  w1[u]  = (rc1[u]  - l1[u]*rp1[u])   / s1[u];   D1[u]  = l1[u]  / s1[u];
            rc2[u]  = s2[u]*l2[u]   - sig_mu;  w2[u]  = (rc2[u]  - l2[u]*rp2[u])   / s2[u];   D2[u]  = l2[u]  / s2[u];
            rc3b[u] = s3b[u]*l3b[u] - sig_mu;  w3b[u] = (rc3b[u] - l3b[u]*rp3b[u]) / s3b[u];  D3b[u] = l3b[u] / s3b[u];
            rc3h[u] = s3h[u]*l3h[u] - sig_mu;  w3h[u] = (rc3h[u] - l3h[u]*rp3h[u]) / s3h[u];  D3h[u] = l3h[u] / s3h[u];
        }
        const float rc3z = s3z*l3z - sig_mu;
        const float w3z  = (rc3z - l3z*rp3z) / s3z;
        const float D3z  = l3z / s3z;

        // ---- arrowhead system M = diag(c) + arrow(alpha, b): pieces + partials ----
        float rhsb[2], rhsh[2], cb[2], ch[2], bbv[2], bhv[2];
        float Tp[2], Ap[2], R3p[2], R4p[2];
        #pragma unroll
        for (int u = 0; u < 2; ++u) {
            const float rdb = qb[u]*zb[u] + pb[u] - l1[u] - l3b[u];   // dual residual, b block
            const float rdh = qh[u]*zh[u] + ph[u] - l2[u] - l3h[u];   // dual residual, h block
            rhsb[u] = -rdb - w1[u] - w3b[u];
            rhsh[u] = -rdh - w2[u] - w3h[u];
            cb[u]  = qb[u] + D1[u] + D3b[u];
            ch[u]  = qh[u] + D2[u] + D3h[u];
            bbv[u] =  D1[u];
            bhv[u] = -D2[u];
            Tp[u]  = l1[u] - l2[u] - w1[u] + w2[u];      // rhs0 coupling sum
            Ap[u]  = D1[u] + D2[u];                       // alpha sum
            R3p[u] = bbv[u]*rhsb[u]/cb[u] + bhv[u]*rhsh[u]/ch[u];
            R4p[u] = bbv[u]*bbv[u]/cb[u]  + bhv[u]*bhv[u]/ch[u];
        }
        const float T  = wave_sum64(Tp[0],  Tp[1]);
        const float SA = wave_sum64(Ap[0],  Ap[1]);
        const float R3 = wave_sum64(R3p[0], R3p[1]);
        const float R4 = wave_sum64(R4p[0], R4p[1]);

        const float rhs0    = -(Qd0*z0 + p0) + T + l3z - w3z;
        const float alpha_m = Qd0 + SA + D3z;
        const float dz0     = (rhs0 - R3) / (alpha_m - R4);   // Schur complement

        // ---- back-substitute, ds, dlam, step-length ratios ----
        float dzb[2], dzh[2], ds1[2], ds2[2], ds3b[2], ds3h[2];
        float dl1[2], dl2[2], dl3b[2], dl3h[2];
        float rmin = big;
        #pragma unroll
        for (int u = 0; u < 2; ++u) {
            dzb[u] = (rhsb[u] - bbv[u]*dz0) / cb[u];
            dzh[u] = (rhsh[u] - bhv[u]*dz0) / ch[u];
            ds1[u]  = -rp1[u]  + dz0 + dzb[u];
            ds2[u]  = -rp2[u]  - dz0 + dzh[u];
            ds3b[u] = -rp3b[u] + dzb[u];
            ds3h[u] = -rp3h[u] + dzh[u];
            dl1[u]  = (-rc1[u]  - l1[u]*ds1[u])   / s1[u];
            dl2[u]  = (-rc2[u]  - l2[u]*ds2[u])   / s2[u];
            dl3b[u] = (-rc3b[u] - l3b[u]*ds3b[u]) / s3b[u];
            dl3h[u] = (-rc3h[u] - l3h[u]*ds3h[u]) / s3h[u];
            rmin = fminf(rmin, stepratio(s1[u],  ds1[u],  big));
            rmin = fminf(rmin, stepratio(s2[u],  ds2[u],  big));
            rmin = fminf(rmin, stepratio(s3b[u], ds3b[u], big));
            rmin = fminf(rmin, stepratio(s3h[u], ds3h[u], big));
            rmin = fminf(rmin, stepratio(l1[u],  dl1[u],  big));
            rmin = fminf(rmin, stepratio(l2[u],  dl2[u],  big));
            rmin = fminf(rmin, stepratio(l3b[u], dl3b[u], big));
            rmin = fminf(rmin, stepratio(l3h[u], dl3h[u], big));
        }
        const float ds3z = -rp3z + dz0;
        const float dl3z = (-rc3z - l3z*ds3z) / s3z;
        rmin = fminf(rmin, stepratio(s3z, ds3z, big));
        rmin = fminf(rmin, stepratio(l3z, dl3z, big));
        rmin = wave_min32(rmin);
        const float a = fminf(1.0f, 0.99f * rmin);

        // ---- update ----
        z0  += a * dz0;
        s3z += a * ds3z;  l3z += a * dl3z;
        #pragma unroll
        for (int u = 0; u < 2; ++u) {
            zb[u]  += a * dzb[u];   zh[u]  += a * dzh[u];
            s1[u]  += a * ds1[u];   s2[u]  += a * ds2[u];
            s3b[u] += a * ds3b[u];  s3h[u] += a * ds3h[u];
            l1[u]  += a * dl1[u];   l2[u]  += a * dl2[u];
            l3b[u] += a * dl3b[u];  l3h[u] += a * dl3h[u];
        }
    }

    if (lane == 0) out[sample] = z0;   // reference returns z[:, :1]
}

extern "C" void kernel_launch(void* const* d_in, const int* in_sizes, int n_in,
                              void* d_out, int out_size, void* d_ws, size_t ws_size,
                              hipStream_t stream) {
    (void)n_in; (void)out_size; (void)d_ws; (void)ws_size;
    const float* y      = (const float*)d_in[0];   // [B, 64]
    const float* q_base = (const float*)d_in[1];   // [129]
    const float* p_base = (const float*)d_in[2];   // [129]
    // d_in[3] = G [257,129] -- never read: its structure IS the algorithm
    const float* h      = (const float*)d_in[4];   // [257]

    const int Bsz = in_sizes[0] / KDIM;            // 512
    const int waves_per_block = 8;                 // 256 threads = 8 wave32
    dim3 block(32 * waves_per_block);
    dim3 grid((Bsz + waves_per_block - 1) / waves_per_block);
    newsvendor_ipm_kernel<<<grid, block, 0, stream>>>(
        y, q_base, p_base, h, (float*)d_out, Bsz);
}